// Router_13761075216758
// MI455X (gfx1250) — compile-verified
//
#include <hip/hip_runtime.h>
#include <math.h>

typedef __attribute__((ext_vector_type(2))) float v2f;
typedef __attribute__((ext_vector_type(4))) float v4f;
typedef __attribute__((ext_vector_type(8))) float v8f;

#define D_MODEL 1024
#define N_EXP 8
#define PAIRS_PER_ROW (D_MODEL / 2)   // 512 float2 pairs per expert row
#define ZPAD_IDX (PAIRS_PER_ROW * N_EXP)

// ---------------------------------------------------------------------------
// Kernel 1: gate scores via V_WMMA_F32_16X16X4_F32, per-token top-2 + softmax.
// One wave32 per 16-token tile. 4 waves / block.
// A (16x4 f32): lane L holds row M=L&15, K = k0 + 2*(L>>4) + {0,1}  (2 VGPRs)
// B (4x16 f32): lane L holds col N=L&15, K = k0 + 2*(L>>4) + {0,1}  (2 VGPRs)
// C/D (16x16 f32): VGPR j, lane L -> M = j + 8*(L>>4), N = L&15
// Lanes with column >= 8 read a zeroed LDS pad with stride 0, so the inner
// loop has no EXEC manipulation: global_load_b64 + ds_load_b64 + v_wmma.
// ---------------------------------------------------------------------------
__global__ __launch_bounds__(128)
void k_gate(const float* __restrict__ x, const float* __restrict__ Wg,
            unsigned* __restrict__ tok_e12, float2* __restrict__ tok_w, int N)
{
    __shared__ v2f   wpair[PAIRS_PER_ROW * N_EXP + 1];  // [pair j][e] + zero pad
    __shared__ float scores[4 * 256];                   // per-wave 16x16 tiles

    const int tid = threadIdx.x;

    // Stage W_gate (8x1024 fp32, 32KB) into LDS in B-fragment pair order:
    // wpair[p*8 + e] = { W[e][2p], W[e][2p+1] }.  Coalesced float2 loads.
    const v2f* Wg2 = (const v2f*)Wg;
    for (int i = tid; i < PAIRS_PER_ROW * N_EXP; i += 128) {
        int e = i >> 9;          // 512 pairs per expert row
        int p = i & 511;
        wpair[p * N_EXP + e] = Wg2[i];
    }
    if (tid == 0) { v2f z = {0.f, 0.f}; wpair[ZPAD_IDX] = z; }
    __syncthreads();

    const int wave = tid >> 5;
    const int lane = tid & 31;
    const int tile = blockIdx.x * 4 + wave;
    const bool active = (tile * 16) < N;   // wave-uniform -> EXEC stays all-1s

    const int h = lane >> 4;     // lane half: K sub-offset 0 or 2
    const int r = lane & 15;     // A: token row in tile; B: expert column

    if (active) {
        const float* xrow = x + (size_t)(tile * 16 + r) * D_MODEL + 2 * h;
        // per-lane B pointer: real fragment for r<8, zero pad (stride 0) else
        const v2f* bptr = (r < N_EXP) ? &wpair[h * N_EXP + r] : &wpair[ZPAD_IDX];
        const int  bstep = (r < N_EXP) ? 2 * N_EXP : 0;  // elements per k0+=4

        v8f acc = {};
        for (int k0 = 0; k0 < D_MODEL; k0 += 4) {
            v2f a = *(const v2f*)(xrow + k0);
            v2f b = *bptr;
            bptr += bstep;
            // D = A(16x4) * B(4x16) + C, fp32 accumulate, RNE
            acc = __builtin_amdgcn_wmma_f32_16x16x4_f32(
                      false, a, false, b, (short)0, acc, false, false);
        }
        // Spill score tile so each token's 8 expert scores land in one lane
#pragma unroll
        for (int j = 0; j < 8; ++j)
            scores[wave * 256 + (j + 8 * h) * 16 + r] = acc[j];
    }
    __syncthreads();

    if (active && lane < 16) {
        const int n = tile * 16 + lane;
        float s[N_EXP];
#pragma unroll
        for (int e = 0; e < N_EXP; ++e)
            s[e] = scores[wave * 256 + lane * 16 + e];

        // top-2 (ties -> lower index, matching jax.lax.top_k)
        int e1 = 0; float v1 = s[0];
#pragma unroll
        for (int e = 1; e < N_EXP; ++e)
            if (s[e] > v1) { v1 = s[e]; e1 = e; }
        int e2 = -1; float v2 = -INFINITY;
#pragma unroll
        for (int e = 0; e < N_EXP; ++e)
            if (e != e1 && s[e] > v2) { v2 = s[e]; e2 = e; }

        // softmax over the two selected scores (others are -inf in reference)
        float ex2 = __expf(v2 - v1);          // v1 >= v2
        float w1  = 1.0f / (1.0f + ex2);
        float w2  = ex2 * w1;

        tok_e12[n] = (unsigned)e1 | ((unsigned)e2 << 8);
        tok_w[n]   = make_float2(w1, w2);
    }
}

// ---------------------------------------------------------------------------
// Kernel 2: deterministic capacity assignment. Single 1024-thread workgroup.
// Per-expert token-order prefix counts packed as 8 x u16 fields in two u64s
// (per-expert total <= N=4096 < 65536 -> no field carry).
// ---------------------------------------------------------------------------
__global__ __launch_bounds__(1024)
void k_scan(const unsigned* __restrict__ tok_e12, int* __restrict__ slots,
            int* __restrict__ Lout, int N, int C)
{
    __shared__ unsigned long long sc0[1024];   // experts 0..3
    __shared__ unsigned long long sc1[1024];   // experts 4..7

    const int t = threadIdx.x;
    const int tpt = (N + 1023) >> 10;          // tokens per thread (4)

    unsigned long long c0 = 0, c1 = 0;
    for (int i = 0; i < tpt; ++i) {
        int n = t * tpt + i;
        if (n < N) {
            unsigned u = tok_e12[n];
            unsigned m = (1u << (u & 255)) | (1u << ((u >> 8) & 255));
#pragma unroll
            for (int e = 0; e < 4; ++e)
                c0 += (unsigned long long)((m >> e) & 1u) << (16 * e);
#pragma unroll
            for (int e = 4; e < 8; ++e)
                c1 += (unsigned long long)((m >> e) & 1u) << (16 * (e - 4));
        }
    }
    sc0[t] = c0; sc1[t] = c1;
    __syncthreads();

    for (int off = 1; off < 1024; off <<= 1) {           // Hillis-Steele
        unsigned long long a0 = (t >= off) ? sc0[t - off] : 0ULL;
        unsigned long long a1 = (t >= off) ? sc1[t - off] : 0ULL;
        __syncthreads();
        sc0[t] += a0; sc1[t] += a1;
        __syncthreads();
    }

    const unsigned long long ex0 = sc0[t] - c0;          // exclusive prefix
    const unsigned long long ex1 = sc1[t] - c1;
    const unsigned long long tot0 = sc0[1023];
    const unsigned long long tot1 = sc1[1023];

    int cnt[N_EXP];
#pragma unroll
    for (int e = 0; e < 4; ++e) cnt[e]     = (int)((ex0 >> (16 * e)) & 0xFFFF);
#pragma unroll
    for (int e = 0; e < 4; ++e) cnt[e + 4] = (int)((ex1 >> (16 * e)) & 0xFFFF);

    for (int i = 0; i < tpt; ++i) {
        int n = t * tpt + i;
        if (n < N) {
            unsigned u = tok_e12[n];
#pragma unroll
            for (int k = 0; k < 2; ++k) {
                int e = (int)((u >> (8 * k)) & 255u);
                int pos = cnt[e]++;
                slots[2 * n + k] = (pos < C) ? pos : C;  // C == dropped sentinel
            }
        }
    }

    if (t == 0) {
#pragma unroll
        for (int e = 0; e < 4; ++e) {
            int t0 = (int)((tot0 >> (16 * e)) & 0xFFFF);
            int t1 = (int)((tot1 >> (16 * e)) & 0xFFFF);
            Lout[e]     = t0 < C ? t0 : C;
            Lout[e + 4] = t1 < C ? t1 : C;
        }
    }
}

// ---------------------------------------------------------------------------
// Kernel 3a: fill final_weights and mask rows. One block per (n,e).
// row[c] = (kept(n,e) && c < L[e]) ? {w_full, 1} : {0, 0}
// v4f non-temporal stores: these 336 MB are write-once, never re-read.
// ---------------------------------------------------------------------------
__global__ __launch_bounds__(256)
void k_rows(const unsigned* __restrict__ tok_e12, const float2* __restrict__ tok_w,
            const int* __restrict__ slots, const int* __restrict__ L,
            float* __restrict__ fw, float* __restrict__ mk, int C)
{
    const int n = blockIdx.x >> 3;
    const int e = blockIdx.x & 7;

    const unsigned u = tok_e12[n];
    const int e1 = (int)(u & 255u), e2 = (int)((u >> 8) & 255u);
    const float2 w = tok_w[n];
    const int s0 = slots[2 * n], s1 = slots[2 * n + 1];

    float vf = 0.f, vm = 0.f;
    if (e == e1 && s0 < C)      { vf = w.x; vm = 1.f; }
    else if (e == e2 && s1 < C) { vf = w.y; vm = 1.f; }

    const int Le = L[e];
    const int C4 = C >> 2;                                // 1280/4, rows 16B-aligned
    v4f* fw4 = (v4f*)(fw + (size_t)blockIdx.x * (size_t)C);
    v4f* mk4 = (v4f*)(mk + (size_t)blockIdx.x * (size_t)C);

    for (int c4 = threadIdx.x; c4 < C4; c4 += 256) {
        const int c = c4 * 4;
        v4f vF, vM;
        vF.x = (c + 0 < Le) ? vf : 0.f;  vM.x = (c + 0 < Le) ? vm : 0.f;
        vF.y = (c + 1 < Le) ? vf : 0.f;  vM.y = (c + 1 < Le) ? vm : 0.f;
        vF.z = (c + 2 < Le) ? vf : 0.f;  vM.z = (c + 2 < Le) ? vm : 0.f;
        vF.w = (c + 3 < Le) ? vf : 0.f;  vM.w = (c + 3 < Le) ? vm : 0.f;
        __builtin_nontemporal_store(vF, fw4 + c4);
        __builtin_nontemporal_store(vM, mk4 + c4);
    }
}

// ---------------------------------------------------------------------------
// Kernel 3b: zero expert_batches rows with slot >= L[e]. Block = one row.
// ---------------------------------------------------------------------------
__global__ __launch_bounds__(256)
void k_ztail(const int* __restrict__ L, float* __restrict__ eb, int C)
{
    const int slot = blockIdx.x;
    const int e    = blockIdx.y;
    if (slot < L[e]) return;   // filled by scatter instead
    v4f z = {0.f, 0.f, 0.f, 0.f};
    v4f* dst = (v4f*)(eb + ((size_t)e * C + slot) * D_MODEL);
    __builtin_nontemporal_store(z, dst + threadIdx.x);  // 256 * 16B = 4KB row
}

// ---------------------------------------------------------------------------
// Kernel 3c: scatter kept token vectors into expert_batches. Block = one (n,k).
// ---------------------------------------------------------------------------
__global__ __launch_bounds__(256)
void k_scatter(const float* __restrict__ x, const unsigned* __restrict__ tok_e12,
               const int* __restrict__ slots, float* __restrict__ eb, int C)
{
    const int nk = blockIdx.x;
    const int slot = slots[nk];
    if (slot >= C) return;     // dropped
    const int n = nk >> 1, k = nk & 1;
    const int e = (int)((tok_e12[n] >> (8 * k)) & 255u);
    const v4f* src = (const v4f*)(x + (size_t)n * D_MODEL);
    v4f* dst = (v4f*)(eb + ((size_t)e * C + slot) * D_MODEL);
    __builtin_nontemporal_store(src[threadIdx.x], dst + threadIdx.x);
}

// ---------------------------------------------------------------------------
extern "C" void kernel_launch(void* const* d_in, const int* in_sizes, int n_in,
                              void* d_out, int out_size, void* d_ws, size_t ws_size,
                              hipStream_t stream)
{
    const float* x  = (const float*)d_in[0];   // (N, 1024) fp32
    const float* Wg = (const float*)d_in[1];   // (8, 1024) fp32

    const int N = in_sizes[0] / D_MODEL;       // 4096
    int C = (int)(2.0 * 1.25 * (double)N / (double)N_EXP);  // floor -> 1280
    C += (C & 1);

    float* out_fw = (float*)d_out;
    float* out_mk = out_fw + (size_t)N * N_EXP * C;
    float* out_eb = out_mk + (size_t)N * N_EXP * C;

    // workspace layout (all fully rewritten every call)
    char* w = (char*)d_ws;
    unsigned* tok_e12 = (unsigned*)w;                               // N u32
    float2*   tok_w   = (float2*)(w + (size_t)N * 4);               // N float2
    int*      slots   = (int*)(w + (size_t)N * 4 + (size_t)N * 8);  // 2N int
    int*      Ldev    = (int*)(w + (size_t)N * 4 + (size_t)N * 8 + (size_t)2 * N * 4);

    // 1) WMMA gating + top2 + softmax
    k_gate<<<(N + 63) / 64, 128, 0, stream>>>(x, Wg, tok_e12, tok_w, N);

    // 2) deterministic capacity scan (single workgroup)
    k_scan<<<1, 1024, 0, stream>>>(tok_e12, slots, Ldev, N, C);

    // 3) output fills (bandwidth-bound: ~380 MB of NT stores)
    k_rows<<<N * N_EXP, 256, 0, stream>>>(tok_e12, tok_w, slots, Ldev,
                                          out_fw, out_mk, C);
    dim3 gz(C, N_EXP);
    k_ztail<<<gz, 256, 0, stream>>>(Ldev, out_eb, C);
    k_scatter<<<2 * N, 256, 0, stream>>>(x, tok_e12, slots, out_eb, C);
}